// GConv_1511828489033
// MI455X (gfx1250) — compile-verified
//
#include <hip/hip_runtime.h>

// ---------------------------------------------------------------------------
// ChebConv (K=6) + InstanceNorm + ReLU for MI455X (gfx1250, wave32).
//   prop: L2-resident f32 atomic scatter, Chebyshev recurrence folded into the
//         scatter (acc init = -T_{k-2}, edge weight scaled by 2).
//   GEMM: V_WMMA_F32_16X16X4_F32 (fp32-native matrix path, 16-row strips).
// ---------------------------------------------------------------------------

typedef float v2f __attribute__((ext_vector_type(2)));
typedef float v8f __attribute__((ext_vector_type(8)));

#define FEAT 64

// ---------------- fill (float4 vectorized, n divisible by 4) ----------------
__global__ void fill_kernel(float4* __restrict__ p, long n4, float v) {
  long i = (long)blockIdx.x * blockDim.x + threadIdx.x;
  long stride = (long)gridDim.x * blockDim.x;
  const float4 vv = make_float4(v, v, v, v);
  for (; i < n4; i += stride) p[i] = vv;
}

// ---------------- dst = -src (float4 vectorized) ----------------
__global__ void negcopy_kernel(float4* __restrict__ dst, const float4* __restrict__ src,
                               long n4) {
  long i = (long)blockIdx.x * blockDim.x + threadIdx.x;
  long stride = (long)gridDim.x * blockDim.x;
  for (; i < n4; i += stride) {
    float4 v = src[i];
    dst[i] = make_float4(-v.x, -v.y, -v.z, -v.w);
  }
}

// ---------------- degree: deg[row[e]] += 1 ----------------
__global__ void deg_kernel(const long long* __restrict__ ei, float* __restrict__ deg, int E) {
  int i = blockIdx.x * blockDim.x + threadIdx.x;
  if (i < E) atomicAdd(&deg[(int)ei[i]], 1.0f);
}

// ---------------- dis[n] = deg>0 ? rsqrt(deg) : 0 ----------------
__global__ void dis_kernel(const float* __restrict__ deg, float* __restrict__ dis, int N) {
  int i = blockIdx.x * blockDim.x + threadIdx.x;
  if (i < N) {
    float d = deg[i];
    dis[i] = (d > 0.0f) ? rsqrtf(d) : 0.0f;
  }
}

// ---------------- prop: acc[row] += scale*dis[row]*dis[col] * t[col] ----------------
// One wave per edge iteration; edge id forced wave-uniform (readfirstlane) so the
// index and dis loads lower to scalar loads; each lane handles 2 contiguous
// features: float2 gather + two global_atomic_add_f32 (L2-resident, 192MB L2).
// scale = -1 gives prop(); scale = -2 gives the 2*prop() of the Chebyshev step.
__global__ void prop_kernel(const float* __restrict__ t, const long long* __restrict__ ei,
                            const float* __restrict__ dis, float* __restrict__ acc,
                            int E, float scale) {
  const int lane = threadIdx.x & 31;
  const int wid  = (int)((blockIdx.x * (long)blockDim.x + threadIdx.x) >> 5);
  const int nw   = (int)(((long)gridDim.x * blockDim.x) >> 5);
  for (int e0 = wid; e0 < E; e0 += nw) {
    const int e = __builtin_amdgcn_readfirstlane(e0);   // wave-uniform -> s_load path
    const int r = (int)ei[e];
    const int c = (int)ei[e + E];
    const float w = scale * dis[r] * dis[c];
    const float2 v = *(const float2*)(t + (long)c * FEAT + lane * 2);
    float* dst = acc + (long)r * FEAT + lane * 2;
    atomicAdd(dst + 0, w * v.x);
    atomicAdd(dst + 1, w * v.y);
  }
}

// ---------------- WMMA GEMM: out (+)= Tx[N,64] @ Wk[64,64] (+ bias on init) ----------
// Per wave: 16-row strip x 64 cols = 4 C-tiles; Fin=64 -> 16 K-steps of K=4.
// A (32-bit 16x4): lane L row M=L%16, VGPR r holds K=2*(L/16)+r -> contiguous float2.
// B (4x16): lane L col n=L%16, VGPR r holds K=2*(L/16)+r -> Wk[(4s+2g+r)*64+n].
// C/D: lane L col n=L%16, VGPR r row M=r+8*(L/16).
__global__ void gemm_wmma(const float* __restrict__ Tx, const float* __restrict__ Wk,
                          const float* __restrict__ bias, float* __restrict__ out,
                          int N, int init) {
  const int lane = threadIdx.x & 31;
  const int wave = threadIdx.x >> 5;
  const int mb = blockIdx.x * 128 + wave * 16;   // 8 waves/block * 16 rows
  if (mb >= N) return;
  const int l = lane & 15;
  const int g = lane >> 4;

  v8f acc[4];
  if (init) {
    for (int nt = 0; nt < 4; ++nt) {
      const float bv = bias[nt * 16 + l];
      for (int r = 0; r < 8; ++r) acc[nt][r] = bv;
    }
  } else {
    for (int nt = 0; nt < 4; ++nt)
      for (int r = 0; r < 8; ++r)
        acc[nt][r] = out[(long)(mb + r + 8 * g) * FEAT + nt * 16 + l];
  }

  const float* arow = Tx + (long)(mb + l) * FEAT + 2 * g;
  for (int s = 0; s < 16; ++s) {
    const float2 av = *(const float2*)(arow + 4 * s);
    v2f a; a[0] = av.x; a[1] = av.y;
    const int k0 = 4 * s + 2 * g;
    for (int nt = 0; nt < 4; ++nt) {
      v2f b;
      b[0] = Wk[(long)k0 * FEAT + nt * 16 + l];
      b[1] = Wk[(long)(k0 + 1) * FEAT + nt * 16 + l];
      acc[nt] = __builtin_amdgcn_wmma_f32_16x16x4_f32(
          /*neg_a=*/false, a, /*neg_b=*/false, b,
          /*c_mod=*/(short)0, acc[nt], /*reuse_a=*/false, /*reuse_b=*/false);
    }
  }

  for (int nt = 0; nt < 4; ++nt)
    for (int r = 0; r < 8; ++r)
      out[(long)(mb + r + 8 * g) * FEAT + nt * 16 + l] = acc[nt][r];
}

// ---------------- InstanceNorm over feature dim + ReLU, one wave per node ----------
__global__ void inorm_relu_kernel(float* __restrict__ out, int N) {
  const int lane = threadIdx.x & 31;
  const int node = (int)((blockIdx.x * (long)blockDim.x + threadIdx.x) >> 5);
  if (node >= N) return;
  float2 v = *(float2*)(out + (long)node * FEAT + lane * 2);
  float s = v.x + v.y;
  float q = v.x * v.x + v.y * v.y;
  for (int m = 16; m >= 1; m >>= 1) {
    s += __shfl_xor(s, m, 32);
    q += __shfl_xor(q, m, 32);
  }
  const float mu  = s * (1.0f / FEAT);
  const float var = q * (1.0f / FEAT) - mu * mu;
  const float inv = rsqrtf(var + 1e-5f);
  v.x = fmaxf((v.x - mu) * inv, 0.0f);
  v.y = fmaxf((v.y - mu) * inv, 0.0f);
  *(float2*)(out + (long)node * FEAT + lane * 2) = v;
}

// ---------------------------------------------------------------------------
extern "C" void kernel_launch(void* const* d_in, const int* in_sizes, int n_in,
                              void* d_out, int out_size, void* d_ws, size_t ws_size,
                              hipStream_t stream) {
  (void)n_in; (void)out_size; (void)ws_size;

  const float*      x   = (const float*)d_in[0];       // [1, N, 64]
  const long long*  ei  = (const long long*)d_in[1];   // [2, E] int64
  const float*      W   = (const float*)d_in[2];       // [K, 64, 64]
  const float*      bb  = (const float*)d_in[3];       // [64]
  float*            out = (float*)d_out;               // [1, N, 64]

  const int N = in_sizes[0] / FEAT;
  const int E = in_sizes[1] / 2;
  const int K = in_sizes[2] / (FEAT * FEAT);
  const long NF = (long)N * FEAT;

  // workspace layout (floats): deg[Npad] | dis[Npad] | bufA[NF] | bufB[NF] | bufC[NF]
  const long Npad = ((long)N + 255) & ~255L;
  float* deg  = (float*)d_ws;
  float* dis  = deg + Npad;
  float* bufA = dis + Npad;
  float* bufB = bufA + NF;
  float* bufC = bufB + NF;

  const int TB = 256;
  const int ewB = 2048;       // blocks for elementwise float4 passes

  // 1) degrees -> dis   (deg fill rounded up to Npad: Npad % 4 == 0)
  hipLaunchKernelGGL(fill_kernel, dim3(128), dim3(TB), 0, stream,
                     (float4*)deg, Npad / 4, 0.0f);
  hipLaunchKernelGGL(deg_kernel, dim3((E + TB - 1) / TB), dim3(TB), 0, stream, ei, deg, E);
  hipLaunchKernelGGL(dis_kernel, dim3((N + TB - 1) / TB), dim3(TB), 0, stream, deg, dis, N);

  const int gemmBlocks = (N + 127) / 128;
  const int propBlocks = 4096;

  // 2) k=0: out = x @ W0 + b
  hipLaunchKernelGGL(gemm_wmma, dim3(gemmBlocks), dim3(TB), 0, stream,
                     x, W, bb, out, N, 1);

  // 3) k=1: T1 = prop(x) in bufB; out += T1 @ W1
  hipLaunchKernelGGL(fill_kernel, dim3(ewB), dim3(TB), 0, stream,
                     (float4*)bufB, NF / 4, 0.0f);
  hipLaunchKernelGGL(prop_kernel, dim3(propBlocks), dim3(TB), 0, stream,
                     x, ei, dis, bufB, E, -1.0f);
  if (K > 1)
    hipLaunchKernelGGL(gemm_wmma, dim3(gemmBlocks), dim3(TB), 0, stream,
                       bufB, W + (long)1 * FEAT * FEAT, bb, out, N, 0);

  // 4) k >= 2: T_k = 2*prop(T_{k-1}) - T_{k-2}, folded into the scatter:
  //    P = -T_{k-2}; P[row] += (-2*dis*dis) * T_{k-1}[col]  ->  P == T_k
  const float* T0 = x;     // T_{k-2}
  float*       T1 = bufB;  // T_{k-1}
  float*       P  = bufC;  // accumulator / next T_k
  for (int k = 2; k < K; ++k) {
    hipLaunchKernelGGL(negcopy_kernel, dim3(ewB), dim3(TB), 0, stream,
                       (float4*)P, (const float4*)T0, NF / 4);
    hipLaunchKernelGGL(prop_kernel, dim3(propBlocks), dim3(TB), 0, stream,
                       T1, ei, dis, P, E, -2.0f);
    hipLaunchKernelGGL(gemm_wmma, dim3(gemmBlocks), dim3(TB), 0, stream,
                       P, W + (long)k * FEAT * FEAT, bb, out, N, 0);
    float* nextP = (T0 == x) ? bufA : (float*)T0;  // rotate 3 buffers
    T0 = T1;
    T1 = P;
    P  = nextP;
  }

  // 5) InstanceNorm + ReLU (one wave32 per node)
  const long inThreads = (long)N * 32;
  hipLaunchKernelGGL(inorm_relu_kernel, dim3((unsigned)((inThreads + TB - 1) / TB)),
                     dim3(TB), 0, stream, out, N);
}